// SEBlock_14757507629900
// MI455X (gfx1250) — compile-verified
//
#include <hip/hip_runtime.h>
#include <math.h>

// Problem constants (reference: x[16,256,128,128] f32)
#define B_   16
#define C_   256
#define CR_  16
#define HW_  16384            // 128*128
#define PLANES_ (B_ * C_)     // 4096

typedef __attribute__((ext_vector_type(2))) float v2f;
typedef __attribute__((ext_vector_type(4))) float v4f;
typedef __attribute__((ext_vector_type(8))) float v8f;

// ---------------------------------------------------------------------------
// Kernel 1: global average pool. One block per (b,c) plane of 16384 floats.
// 256 threads * 16 float4 each = 16384 floats. LDS tree reduction.
// ---------------------------------------------------------------------------
__global__ void se_pool_kernel(const float* __restrict__ x,
                               float* __restrict__ pooled) {
    const int plane = blockIdx.x;                  // 0..4095 == b*256 + c
    const v4f* __restrict__ px =
        (const v4f*)(x + (size_t)plane * HW_);     // 4096 float4s
    const int t = threadIdx.x;                     // 0..255

    float sum = 0.0f;
#pragma unroll
    for (int i = 0; i < 16; ++i) {
        v4f v = px[t + i * 256];
        sum += (v.x + v.y) + (v.z + v.w);
    }

    __shared__ float red[256];
    red[t] = sum;
    __syncthreads();
#pragma unroll
    for (int s = 128; s > 0; s >>= 1) {
        if (t < s) red[t] += red[t + s];
        __syncthreads();
    }
    if (t == 0) pooled[plane] = red[0] * (1.0f / (float)HW_);
}

// ---------------------------------------------------------------------------
// Kernel 2: the SE gate — two tiny GEMMs done with V_WMMA_F32_16X16X4_F32.
// Single wave (32 lanes), EXEC all ones, uniform control flow.
//
// f32 16x16x4 operand layouts (CDNA5 ISA 7.12.2):
//   A (16x4, MxK), 2 VGPRs: lanes 0-15 -> M=lane, VGPR0=K0, VGPR1=K1;
//                           lanes 16-31 -> M=lane-16, VGPR0=K2, VGPR1=K3.
//   B (4x16, KxN), 2 VGPRs: mirrored (N across lanes, K split the same way).
//   C/D (16x16), 8 VGPRs:   VGPR j: lanes 0-15 -> M=j, lanes 16-31 -> M=j+8,
//                           N = lane%16.
// ---------------------------------------------------------------------------
__global__ void se_gate_kernel(const float* __restrict__ pooled,  // [16,256]
                               const float* __restrict__ w1,      // [16,256]
                               const float* __restrict__ b1,      // [16]
                               const float* __restrict__ w2,      // [256,16]
                               const float* __restrict__ b2,      // [256]
                               float* __restrict__ s_out) {       // [16,256]
    const int lane = threadIdx.x;      // 0..31
    const int half = lane >> 4;        // 0 / 1
    const int l    = lane & 15;        // 0..15

    // ---- GEMM1: h[b,r] = sum_k pooled[b,k] * w1[r,k]   (M=16,N=16,K=256)
    v8f acc = {};
    for (int k0 = 0; k0 < C_; k0 += 4) {
        const int k = k0 + 2 * half;   // this half-wave's K pair
        v2f a, b;
        a.x = pooled[l * C_ + k];      // A[M=l][k]
        a.y = pooled[l * C_ + k + 1];  // A[M=l][k+1]
        b.x = w1[l * C_ + k];          // B[k][N=l]   = w1[l][k]
        b.y = w1[l * C_ + k + 1];      // B[k+1][N=l]
        acc = __builtin_amdgcn_wmma_f32_16x16x4_f32(
            /*neg_a=*/false, a, /*neg_b=*/false, b,
            /*c_mod=*/(short)0, acc, /*reuse_a=*/false, /*reuse_b=*/false);
    }

    // bias + LeakyReLU(0.01), stash h row-major [M][r] in LDS
    __shared__ float hbuf[CR_ * CR_];  // 16x16
    {
        const float bias = b1[l];      // N = r = l
#pragma unroll
        for (int j = 0; j < 8; ++j) {
            float v = acc[j] + bias;
            v = (v >= 0.0f) ? v : 0.01f * v;
            const int M = j + 8 * half;
            hbuf[M * CR_ + l] = v;
        }
    }
    __syncthreads();

    // ---- GEMM2: s_pre[b,c] = sum_r h[b,r] * w2[c,r]    (M=16,K=16,N=256)
    for (int tile = 0; tile < C_ / 16; ++tile) {
        v8f acc2 = {};
#pragma unroll
        for (int k0 = 0; k0 < CR_; k0 += 4) {
            const int k = k0 + 2 * half;
            v2f a, b;
            a.x = hbuf[l * CR_ + k];                  // A[M=l][k]
            a.y = hbuf[l * CR_ + k + 1];
            const int n = tile * 16 + l;              // output channel
            b.x = w2[n * CR_ + k];                    // B[k][n] = w2[n][k]
            b.y = w2[n * CR_ + k + 1];
            acc2 = __builtin_amdgcn_wmma_f32_16x16x4_f32(
                false, a, false, b, (short)0, acc2, false, false);
        }
        // bias + sigmoid, scatter to s_out[b, n]
        const int n = tile * 16 + l;
        const float bias = b2[n];
#pragma unroll
        for (int j = 0; j < 8; ++j) {
            const int M = j + 8 * half;
            float v = acc2[j] + bias;
            float sg = 1.0f / (1.0f + __expf(-v));
            s_out[M * C_ + n] = sg;
        }
    }
}

// ---------------------------------------------------------------------------
// Kernel 3: out = x * s[b,c].  One block per plane, float4, nontemporal
// stores (output is never re-read; keep L2 for the x stream).
// ---------------------------------------------------------------------------
__global__ void se_scale_kernel(const float* __restrict__ x,
                                const float* __restrict__ s,
                                float* __restrict__ out) {
    const int plane = blockIdx.x;
    const float scale = s[plane];
    const v4f* __restrict__ px = (const v4f*)(x + (size_t)plane * HW_);
    v4f* __restrict__ po = (v4f*)(out + (size_t)plane * HW_);
    const int t = threadIdx.x;
#pragma unroll
    for (int i = 0; i < 16; ++i) {
        v4f v = px[t + i * 256];
        v = v * scale;
        __builtin_nontemporal_store(v, &po[t + i * 256]);
    }
}

// ---------------------------------------------------------------------------
extern "C" void kernel_launch(void* const* d_in, const int* in_sizes, int n_in,
                              void* d_out, int out_size, void* d_ws, size_t ws_size,
                              hipStream_t stream) {
    const float* x  = (const float*)d_in[0];
    const float* w1 = (const float*)d_in[1];
    const float* b1 = (const float*)d_in[2];
    const float* w2 = (const float*)d_in[3];
    const float* b2 = (const float*)d_in[4];
    float* out = (float*)d_out;

    float* pooled = (float*)d_ws;          // 4096 floats
    float* sgate  = pooled + PLANES_;      // 4096 floats

    se_pool_kernel<<<PLANES_, 256, 0, stream>>>(x, pooled);
    se_gate_kernel<<<1, 32, 0, stream>>>(pooled, w1, b1, w2, b2, sgate);
    se_scale_kernel<<<PLANES_, 256, 0, stream>>>(x, sgate, out);
}